// RankingLossL1_65970697666792
// MI455X (gfx1250) — compile-verified
//
#include <hip/hip_runtime.h>
#include <float.h>

#define N_PTS   16384
#define DIMS    128
#define T_ANCH  2048
#define KTOP    10
#define GAMMA_F 1.0f

#define CT      128     // candidates staged per tile (4 per lane)
#define AW      8       // anchors (one per wave) per workgroup
#define BPAD    132     // padded LDS row stride in floats (528B = 33*16B, bank-rotating)

// ---- CDNA5 async global->LDS path (guarded; falls back to sync copy) ----
typedef int v4i __attribute__((ext_vector_type(4)));
typedef __attribute__((address_space(1))) v4i g_v4i;   // global int4
typedef __attribute__((address_space(3))) v4i l_v4i;   // LDS int4

#if __has_builtin(__builtin_amdgcn_global_load_async_to_lds_b128) && \
    __has_builtin(__builtin_amdgcn_s_wait_asynccnt)
#define USE_ASYNC 1
#else
#define USE_ASYNC 0
#endif

__device__ __forceinline__ void stage_b128(const float* src, float* ldst) {
#if USE_ASYNC
  __builtin_amdgcn_global_load_async_to_lds_b128(
      (g_v4i*)(void*)(const_cast<float*>(src)),
      (l_v4i*)(void*)(ldst), 0, 0);
#else
  *(float4*)ldst = *(const float4*)src;
#endif
}

__device__ __forceinline__ void wait_stage() {
#if USE_ASYNC
  __builtin_amdgcn_s_wait_asynccnt(0);
#endif
}

// ============================================================================
// Kernel 1: fused L1-cdist + exact top-K.  gridDim = (T/AW, 2 directions).
// 256 threads = 8 wave32s; wave w owns anchor blockIdx.x*8+w; each lane
// processes 4 candidates per tile, sharing the wave-broadcast anchor float4.
// ============================================================================
__global__ __launch_bounds__(256, 4) void cdist_topk_kernel(
    const float* __restrict__ out1, const float* __restrict__ out2,
    const int*   __restrict__ anchor1, const int* __restrict__ anchor2,
    int* __restrict__ negout)                    // [2][T_ANCH*KTOP]
{
  __shared__ __align__(16) float As[AW][BPAD];
  __shared__ __align__(16) float Bs[2][CT][BPAD];
  __shared__ float Md[AW][32 * KTOP];
  __shared__ int   Mi[AW][32 * KTOP];

  const int tid  = threadIdx.x;
  const int lane = tid & 31;
  const int w    = tid >> 5;
  const int dir  = blockIdx.y;

  const float* __restrict__ Asrc = dir ? out2 : out1;
  const float* __restrict__ Bsrc = dir ? out1 : out2;
  const int*   __restrict__ anch = dir ? anchor2 : anchor1;

  // ---- load 8 anchor vectors into LDS (256 threads x one float4 each) ----
  {
    const int row = tid >> 5;             // 0..7
    const int c4  = (tid & 31) * 4;       // 0..124
    const int t   = blockIdx.x * AW + row;
    const float4 a = *(const float4*)(Asrc + (size_t)anch[t] * DIMS + c4);
    As[row][c4 + 0] = a.x; As[row][c4 + 1] = a.y;
    As[row][c4 + 2] = a.z; As[row][c4 + 3] = a.w;
  }

  // ---- per-lane exact top-K registers ----
  float bd[KTOP];
  int   bi[KTOP];
#pragma unroll
  for (int j = 0; j < KTOP; ++j) { bd[j] = FLT_MAX; bi[j] = 0; }

  auto insert_topk = [&](float d, int idx) {
    if (d < bd[KTOP - 1]) {
      bd[KTOP - 1] = d; bi[KTOP - 1] = idx;
#pragma unroll
      for (int j = KTOP - 1; j > 0; --j)
        if (bd[j] < bd[j - 1]) {
          float td = bd[j]; bd[j] = bd[j - 1]; bd[j - 1] = td;
          int   ti = bi[j]; bi[j] = bi[j - 1]; bi[j - 1] = ti;
        }
    }
  };

  // ---- tile staging: CT*DIMS/4 = 4096 float4 chunks, 16 per thread ----
  auto stage_tile = [&](int tile, int buf) {
    const float* src = Bsrc + (size_t)tile * CT * DIMS;
#pragma unroll
    for (int i = 0; i < (CT * DIMS / 4) / 256; ++i) {
      const int q   = tid + i * 256;
      const int row = q >> 5;             // 0..127
      const int col = (q & 31) * 4;       // 0..124
      stage_b128(src + row * DIMS + col, &Bs[buf][row][col]);
    }
  };

  stage_tile(0, 0);

  const int NT = N_PTS / CT;              // 128 tiles
  for (int tile = 0; tile < NT; ++tile) {
    const int buf = tile & 1;
    wait_stage();
    __syncthreads();                      // tile data visible; prev compute done
    if (tile + 1 < NT) stage_tile(tile + 1, buf ^ 1);

    // lane handles candidates lane + {0,32,64,96} of this tile.
    const float* __restrict__ brow0 = &Bs[buf][lane +  0][0];
    const float* __restrict__ brow1 = &Bs[buf][lane + 32][0];
    const float* __restrict__ brow2 = &Bs[buf][lane + 64][0];
    const float* __restrict__ brow3 = &Bs[buf][lane + 96][0];
    const float* __restrict__ arow  = &As[w][0];
    float4 s0 = {0.f,0.f,0.f,0.f}, s1 = {0.f,0.f,0.f,0.f};
    float4 s2 = {0.f,0.f,0.f,0.f}, s3 = {0.f,0.f,0.f,0.f};
#pragma unroll 4
    for (int d = 0; d < DIMS; d += 4) {
      const float4 a  = *(const float4*)(arow  + d);
      const float4 b0 = *(const float4*)(brow0 + d);
      const float4 b1 = *(const float4*)(brow1 + d);
      const float4 b2 = *(const float4*)(brow2 + d);
      const float4 b3 = *(const float4*)(brow3 + d);
      s0.x += __builtin_fabsf(a.x - b0.x); s0.y += __builtin_fabsf(a.y - b0.y);
      s0.z += __builtin_fabsf(a.z - b0.z); s0.w += __builtin_fabsf(a.w - b0.w);
      s1.x += __builtin_fabsf(a.x - b1.x); s1.y += __builtin_fabsf(a.y - b1.y);
      s1.z += __builtin_fabsf(a.z - b1.z); s1.w += __builtin_fabsf(a.w - b1.w);
      s2.x += __builtin_fabsf(a.x - b2.x); s2.y += __builtin_fabsf(a.y - b2.y);
      s2.z += __builtin_fabsf(a.z - b2.z); s2.w += __builtin_fabsf(a.w - b2.w);
      s3.x += __builtin_fabsf(a.x - b3.x); s3.y += __builtin_fabsf(a.y - b3.y);
      s3.z += __builtin_fabsf(a.z - b3.z); s3.w += __builtin_fabsf(a.w - b3.w);
    }
    const int base = tile * CT + lane;
    insert_topk((s0.x + s0.y) + (s0.z + s0.w), base +  0);
    insert_topk((s1.x + s1.y) + (s1.z + s1.w), base + 32);
    insert_topk((s2.x + s2.y) + (s2.z + s2.w), base + 64);
    insert_topk((s3.x + s3.y) + (s3.z + s3.w), base + 96);
  }

  // ---- merge 32 sorted 10-lists -> global top-10 for this wave's anchor ----
#pragma unroll
  for (int j = 0; j < KTOP; ++j) {
    Md[w][lane * KTOP + j] = bd[j];
    Mi[w][lane * KTOP + j] = bi[j];
  }
  __syncthreads();

  int p = 0;
  const int tA = blockIdx.x * AW + w;
  for (int k = 0; k < KTOP; ++k) {
    float dmin = (p < KTOP) ? Md[w][lane * KTOP + p] : FLT_MAX;
    int   wl   = lane;
#pragma unroll
    for (int off = 16; off > 0; off >>= 1) {
      const float od = __shfl_xor(dmin, off, 32);
      const int   ol = __shfl_xor(wl,   off, 32);
      if (od < dmin || (od == dmin && ol < wl)) { dmin = od; wl = ol; }
    }
    const int pw = __shfl(p, wl, 32);
    if (lane == 0)
      negout[(size_t)dir * (T_ANCH * KTOP) + tA * KTOP + k] = Mi[w][wl * KTOP + pw];
    if (lane == wl) ++p;
  }
}

// ============================================================================
// Kernel 2: hinge-loss reduction over T*K terms -> scalar
// ============================================================================
__global__ __launch_bounds__(256, 4) void loss_kernel(
    const float* __restrict__ out1, const float* __restrict__ out2,
    const int*   __restrict__ anchor1, const int* __restrict__ anchor2,
    const int*   __restrict__ neg, float* __restrict__ out)
{
  const int j = blockIdx.x * blockDim.x + threadIdx.x;
  float term = 0.f;
  if (j < T_ANCH * KTOP) {
    const int t = j / KTOP;
    const float* av1 = out1 + (size_t)anchor1[t] * DIMS;
    const float* av2 = out2 + (size_t)anchor2[t] * DIMS;
    const float* nv1 = out2 + (size_t)neg[j] * DIMS;
    const float* nv2 = out1 + (size_t)neg[T_ANCH * KTOP + j] * DIMS;
    float pos = 0.f, dn1 = 0.f, dn2 = 0.f;
#pragma unroll 4
    for (int d = 0; d < DIMS; d += 4) {
      const float4 a1 = *(const float4*)(av1 + d);
      const float4 a2 = *(const float4*)(av2 + d);
      const float4 n1 = *(const float4*)(nv1 + d);
      const float4 n2 = *(const float4*)(nv2 + d);
      pos += __builtin_fabsf(a1.x - a2.x) + __builtin_fabsf(a1.y - a2.y)
           + __builtin_fabsf(a1.z - a2.z) + __builtin_fabsf(a1.w - a2.w);
      dn1 += __builtin_fabsf(a1.x - n1.x) + __builtin_fabsf(a1.y - n1.y)
           + __builtin_fabsf(a1.z - n1.z) + __builtin_fabsf(a1.w - n1.w);
      dn2 += __builtin_fabsf(a2.x - n2.x) + __builtin_fabsf(a2.y - n2.y)
           + __builtin_fabsf(a2.z - n2.z) + __builtin_fabsf(a2.w - n2.w);
    }
    const float Dm = pos + GAMMA_F;
    term = fmaxf(Dm - dn1, 0.f) + fmaxf(Dm - dn2, 0.f);
  }
  __shared__ float red[256];
  red[threadIdx.x] = term;
  __syncthreads();
#pragma unroll
  for (int s = 128; s > 0; s >>= 1) {
    if (threadIdx.x < s) red[threadIdx.x] += red[threadIdx.x + s];
    __syncthreads();
  }
  if (threadIdx.x == 0)
    atomicAdd(out, red[0] * (1.0f / (float)(T_ANCH * KTOP)));
}

__global__ void zero_kernel(float* out) { out[0] = 0.f; }

// ============================================================================
extern "C" void kernel_launch(void* const* d_in, const int* in_sizes, int n_in,
                              void* d_out, int out_size, void* d_ws, size_t ws_size,
                              hipStream_t stream) {
  const float* out1    = (const float*)d_in[0];
  const float* out2    = (const float*)d_in[1];
  const int*   anchor1 = (const int*)d_in[2];
  const int*   anchor2 = (const int*)d_in[3];
  int*   neg = (int*)d_ws;                 // [2][T_ANCH*KTOP] int32
  float* out = (float*)d_out;

  zero_kernel<<<1, 1, 0, stream>>>(out);

  dim3 grid(T_ANCH / AW, 2);
  cdist_topk_kernel<<<grid, 256, 0, stream>>>(out1, out2, anchor1, anchor2, neg);

  loss_kernel<<<(T_ANCH * KTOP + 255) / 256, 256, 0, stream>>>(
      out1, out2, anchor1, anchor2, neg, out);
}